// SingleHashNMSKPtC_71554155151811
// MI455X (gfx1250) — compile-verified
//
#include <hip/hip_runtime.h>
#include <stdint.h>

// Hashed single-pass NMS on MI455X (gfx1250, wave32).
// Streaming passes at HBM rate + L2-resident open-addressed hash table.
// Slot layout: interleaved 16B {key, payload} so CAS/max/reads share a cacheline.

#define EMPTY_KEY 0xFFFFFFFFFFFFFFFFULL
#define L2A      (-0.4941092f)   // log2(0.71)
#define INV_L2A  (-2.0238445f)   // 1/log2(0.71)

__device__ __forceinline__ uint64_t mix64(uint64_t x) {
    // MurmurHash3 fmix64 finalizer
    x ^= x >> 33; x *= 0xff51afd7ed558ccdULL;
    x ^= x >> 33; x *= 0xc4ceb9fe1a85ec53ULL;
    x ^= x >> 33;
    return x;
}

__device__ __forceinline__ uint64_t box_key(float cx, float cy, float w, float h) {
    // iw = round(log(w/8)/log(0.71)) in log2 domain -> v_log_f32 / v_exp_f32
    float iw = floorf(__log2f(w * 0.125f) * INV_L2A + 0.5f);
    float ih = floorf(__log2f(h * 0.125f) * INV_L2A + 0.5f);
    float dw = 8.0f * __builtin_amdgcn_exp2f(iw * L2A);
    float dh = 8.0f * __builtin_amdgcn_exp2f(ih * L2A);
    float ix = floorf(__fdividef(cx, dw) + 0.5f);
    float iy = floorf(__fdividef(cy, dh) + 0.5f);
    uint64_t qiw = (uint64_t)((long long)iw) & 0xFFFFULL;
    uint64_t qih = (uint64_t)((long long)ih) & 0xFFFFULL;
    uint64_t qix = (uint64_t)((long long)ix) & 0xFFFFULL;
    uint64_t qiy = (uint64_t)((long long)iy) & 0xFFFFULL;
    return (qiw << 48) | (qih << 32) | (qix << 16) | qiy;
}

// ---- pass 0: init table (b128 stores, one per 16B slot) ----------------------
__global__ __launch_bounds__(256) void nms_init_kernel(ulonglong2* __restrict__ tab,
                                                       uint32_t M) {
    uint32_t i = blockIdx.x * blockDim.x + threadIdx.x;
    uint32_t stride = gridDim.x * blockDim.x;
    ulonglong2 empty;
    empty.x = EMPTY_KEY;
    empty.y = 0ULL;               // any real payload has high ~idx bits -> > 0
    for (; i < M; i += stride) tab[i] = empty;   // global_store_b128
}

// ---- pass 1: claim slot + atomicMax packed (conf, ~idx) ----------------------
__global__ __launch_bounds__(256) void nms_insert_kernel(const float4* __restrict__ rects,
                                                         const float*  __restrict__ conf,
                                                         uint64_t* __restrict__ tab,
                                                         uint32_t n, uint32_t M) {
    uint32_t i = blockIdx.x * blockDim.x + threadIdx.x;
    if (i >= n) return;

    if (i + 4096u < n) __builtin_prefetch((const void*)(rects + i + 4096u), 0, 1);

    float4 r = rects[i];                     // global_load_b128, coalesced
    uint64_t key = box_key(r.x, r.y, r.z, r.w);

    uint32_t cb = __float_as_uint(conf[i]);  // conf in [0,1): bits monotone
    uint64_t packed = ((uint64_t)cb << 32) | (uint64_t)(~i); // max conf, min idx

    uint32_t mask = M - 1u;
    uint32_t hslot = (uint32_t)mix64(key) & mask;
    for (uint32_t p = 0; p < M; ++p) {       // capped linear probing (no hangs)
        uint64_t* kp = &tab[(size_t)hslot * 2u];
        uint64_t k = *kp;                    // cheap filter: keys are write-once,
        if (k == EMPTY_KEY) {                // a stale view can only be EMPTY
            k = (uint64_t)atomicCAS((unsigned long long*)kp,
                                    (unsigned long long)EMPTY_KEY,
                                    (unsigned long long)key);
            if (k == EMPTY_KEY || k == key) {
                atomicMax((unsigned long long*)(kp + 1), (unsigned long long)packed);
                return;
            }
        } else if (k == key) {               // common case: slot already claimed
            atomicMax((unsigned long long*)(kp + 1), (unsigned long long)packed);
            return;
        }
        hslot = (hslot + 1u) & mask;
    }
}

// ---- pass 2: re-probe (b128 slot loads), decide winner, write outputs --------
__global__ __launch_bounds__(256) void nms_resolve_kernel(const float4* __restrict__ rects,
                                                          const float*  __restrict__ conf,
                                                          const ulonglong2* __restrict__ tab,
                                                          float* __restrict__ out,
                                                          uint32_t n, uint32_t M) {
    uint32_t i = blockIdx.x * blockDim.x + threadIdx.x;
    if (i >= n) return;

    if (i + 4096u < n) __builtin_prefetch((const void*)(rects + i + 4096u), 0, 1);

    float4 r = rects[i];
    uint64_t key = box_key(r.x, r.y, r.z, r.w);

    float c = conf[i];
    uint64_t packed = ((uint64_t)__float_as_uint(c) << 32) | (uint64_t)(~i);

    uint32_t mask = M - 1u;
    uint32_t hslot = (uint32_t)mix64(key) & mask;
    bool keep = true;                         // fail-open if probe exhausts
    for (uint32_t p = 0; p < M; ++p) {
        ulonglong2 slot = tab[hslot];         // one global_load_b128 per probe
        if (slot.x == key)      { keep = (slot.y == packed); break; }
        if (slot.x == EMPTY_KEY) break;       // shouldn't happen; fail-open
        hslot = (hslot + 1u) & mask;
    }

    out[i]     = keep ? 1.0f : 0.0f;          // keep mask
    out[n + i] = keep ? c : 0.0f;             // suppressed conf
}

extern "C" void kernel_launch(void* const* d_in, const int* in_sizes, int n_in,
                              void* d_out, int out_size, void* d_ws, size_t ws_size,
                              hipStream_t stream) {
    const float4* rects = (const float4*)d_in[0];   // (N,4) f32 -> float4 loads
    const float*  conf  = (const float*)d_in[1];    // (N,)  f32
    float* out = (float*)d_out;                     // [keep(0/1) | kept_conf]
    uint32_t n = (uint32_t)in_sizes[1];

    // 16 bytes/slot, power-of-two M sized to workspace, capped at 2^21 slots
    // (32 MB -> fully resident in the 192 MB L2).
    uint32_t M = 1u << 21;
    while ((size_t)M * 16u > ws_size && M > (1u << 16)) M >>= 1;
    ulonglong2* tab = (ulonglong2*)d_ws;

    const int tb = 256;
    int init_blocks = (int)((M + tb - 1) / tb);
    nms_init_kernel<<<init_blocks, tb, 0, stream>>>(tab, M);

    int blocks = (int)((n + tb - 1) / tb);
    nms_insert_kernel<<<blocks, tb, 0, stream>>>(rects, conf, (uint64_t*)tab, n, M);
    nms_resolve_kernel<<<blocks, tb, 0, stream>>>(rects, conf, tab, out, n, M);
}